// ExpectedCalibrationError_52991306498503
// MI455X (gfx1250) — compile-verified
//
#include <hip/hip_runtime.h>
#include <hip/hip_bf16.h>
#include <stdint.h>
#include <math.h>

#define N_BINS 15
#define BLOCK  256
#define WPB    (BLOCK / 32)   // waves per block (wave32)
#define MAXCH  26             // max float4 chunks per row supported (C <= 104)

// ---- CDNA5 async global->LDS path (guarded; fallback = direct loads) ----
#if __has_builtin(__builtin_amdgcn_global_load_async_to_lds_b128)
#define USE_ASYNC 1
#else
#define USE_ASYNC 0
#endif

#if USE_ASYNC
typedef int v4i_t __attribute__((vector_size(16)));

__device__ __forceinline__ void async_copy_b128(const void* gsrc, void* ldst) {
  // global_load_async_to_lds_b128: copies 16 bytes per lane from global gsrc to LDS ldst
  __builtin_amdgcn_global_load_async_to_lds_b128(
      (__attribute__((address_space(1))) v4i_t*)(uintptr_t)gsrc,
      (__attribute__((address_space(3))) v4i_t*)(uint32_t)(uintptr_t)ldst,
      /*offset=*/0, /*cpol=*/0);
}
__device__ __forceinline__ void wait_async_le1() {
#if __has_builtin(__builtin_amdgcn_s_wait_asynccnt)
  __builtin_amdgcn_s_wait_asynccnt(1);
  asm volatile("" ::: "memory");
#else
  asm volatile("s_wait_asynccnt 1" ::: "memory");
#endif
}
__device__ __forceinline__ void wait_async_le0() {
#if __has_builtin(__builtin_amdgcn_s_wait_asynccnt)
  __builtin_amdgcn_s_wait_asynccnt(0);
  asm volatile("" ::: "memory");
#else
  asm volatile("s_wait_asynccnt 0" ::: "memory");
#endif
}
#endif

__global__ __launch_bounds__(BLOCK)
void ece_main_kernel(const float* __restrict__ logits,
                     const int* __restrict__ labels,
                     float* __restrict__ ws, int n, int c) {
  __shared__ float sbins[3 * N_BINS];
#if USE_ASYNC
  __shared__ float4 stage[WPB][2][MAXCH];
#endif
  const int tid  = threadIdx.x;
  const int lane = tid & 31;
  const int wv   = tid >> 5;

  for (int i = tid; i < 3 * N_BINS; i += BLOCK) sbins[i] = 0.0f;
  __syncthreads();

  const int  chunks = (c + 3) >> 2;     // float4 chunks per row (25 for C=100)
  const bool al     = (lane < chunks);  // lane owns a chunk
  const int  gw     = blockIdx.x * WPB + wv;   // global wave id = first row
  const int  nw     = gridDim.x * WPB;         // total waves = row stride

#if USE_ASYNC
  int buf = 0;
  if (gw < n && al) {
    const float4* g = (const float4*)(logits + (size_t)gw * c) + lane;
    async_copy_b128(g, &stage[wv][0][lane]);
  }
#endif

  for (int row = gw; row < n; row += nw) {
    const int nxt = row + nw;
    float4 v = make_float4(-INFINITY, -INFINITY, -INFINITY, -INFINITY);

#if USE_ASYNC
    if (nxt < n) {
      if (al) {
        const float4* g = (const float4*)(logits + (size_t)nxt * c) + lane;
        async_copy_b128(g, &stage[wv][buf ^ 1][lane]);
      }
      wait_async_le1();   // current buffer complete (loads retire in order)
    } else {
      wait_async_le0();   // last iteration: drain
    }
    if (al) v = stage[wv][buf][lane];
    buf ^= 1;
#else
    if (nxt < n && al)
      __builtin_prefetch(logits + (size_t)nxt * c + lane * 4, 0, 0);
    if (al) v = ((const float4*)(logits + (size_t)row * c))[lane];
#endif

    // local max/argmax over this lane's 4 elements (first-index wins ties)
    const int base = lane << 2;
    float bv = v.x; int bi = base;
    if (v.y > bv) { bv = v.y; bi = base + 1; }
    if (v.z > bv) { bv = v.z; bi = base + 2; }
    if (v.w > bv) { bv = v.w; bi = base + 3; }

    // wave32 argmax reduction, smaller index wins ties
    for (int off = 16; off; off >>= 1) {
      float ov = __shfl_xor(bv, off, 32);
      int   oi = __shfl_xor(bi, off, 32);
      if (ov > bv || (ov == bv && oi < bi)) { bv = ov; bi = oi; }
    }

    // sum of exp(x - max); padded lanes hold -inf -> exp = 0
    float s = __expf(v.x - bv) + __expf(v.y - bv) +
              __expf(v.z - bv) + __expf(v.w - bv);
    for (int off = 16; off; off >>= 1) s += __shfl_xor(s, off, 32);

    if (lane == 0) {
      const float conf = 1.0f / s;                    // == max(softmax(row))
      const float acc  = (bi == labels[row]) ? 1.0f : 0.0f;
      int b = (int)ceilf(conf * (float)N_BINS) - 1;
      b = b < 0 ? 0 : (b > N_BINS - 1 ? N_BINS - 1 : b);
      atomicAdd(&sbins[b], 1.0f);                     // ds_add_f32
      atomicAdd(&sbins[N_BINS + b], conf);
      atomicAdd(&sbins[2 * N_BINS + b], acc);
    }
  }

  __syncthreads();
  for (int i = tid; i < 3 * N_BINS; i += BLOCK)
    atomicAdd(&ws[i], sbins[i]);                      // global_atomic_add_f32
}

__global__ void ece_zero_kernel(float* __restrict__ ws) {
  if (threadIdx.x < 3 * N_BINS) ws[threadIdx.x] = 0.0f;
}

__global__ void ece_final_kernel(const float* __restrict__ ws,
                                 float* __restrict__ out, int n) {
  const int lane = threadIdx.x & 31;
  float pb = 0.0f;
  if (lane < N_BINS) {
    const float cnt = ws[lane];
    const float cs  = ws[N_BINS + lane];
    const float as  = ws[2 * N_BINS + lane];
    if (cnt > 0.0f) {
      const float denom = fmaxf(cnt, 1.0f);
      pb = fabsf(cs / denom - as / denom) * (cnt / (float)n);
    }
  }
  for (int off = 16; off; off >>= 1) pb += __shfl_xor(pb, off, 32);
  if (lane == 0) out[0] = pb;
}

extern "C" void kernel_launch(void* const* d_in, const int* in_sizes, int n_in,
                              void* d_out, int out_size, void* d_ws, size_t ws_size,
                              hipStream_t stream) {
  const float* logits = (const float*)d_in[0];
  const int*   labels = (const int*)d_in[1];
  float* ws  = (float*)d_ws;
  float* out = (float*)d_out;

  const int n = in_sizes[1];              // 1,000,000 rows
  const int c = in_sizes[0] / n;          // 100 classes

  ece_zero_kernel<<<1, 64, 0, stream>>>(ws);

  int blocks = 2048;                      // 16384 waves; ~61 rows per wave
  const int max_blocks = (n + WPB - 1) / WPB;
  if (blocks > max_blocks) blocks = max_blocks;
  if (blocks < 1) blocks = 1;

  ece_main_kernel<<<blocks, BLOCK, 0, stream>>>(logits, labels, ws, n, c);
  ece_final_kernel<<<1, 32, 0, stream>>>(ws, out, n);
}